// Gemma4AssistantMaskedEmbedder_47029891891428
// MI455X (gfx1250) — compile-verified
//
#include <hip/hip_runtime.h>
#include <hip/hip_bf16.h>

typedef __attribute__((ext_vector_type(2))) float v2f;
typedef __attribute__((ext_vector_type(8))) float v8f;

#define H_DIM 1024   // hidden size fixed by the reference

// ---- order-preserving float <-> uint encoding (for atomic min) -------------
__device__ __forceinline__ unsigned fenc(float f) {
  unsigned u = __float_as_uint(f);
  return (u & 0x80000000u) ? ~u : (u | 0x80000000u);
}
__device__ __forceinline__ float fdec(unsigned u) {
  u = (u & 0x80000000u) ? (u & 0x7fffffffu) : ~u;
  return __uint_as_float(u);
}

__global__ void init_min_kernel(unsigned* __restrict__ minEnc) {
  *minEnc = fenc(__int_as_float(0x7f800000)); // +inf
}

// ---- Phase 1: centroid logits GEMM via V_WMMA_F32_16X16X4_F32 --------------
// One wave (32 threads) per 16x16 output tile; K loop over H in steps of 4.
// A-frag: lane<16 -> {X[t][k0],X[t][k0+1]}, lane>=16 -> {X[t][k0+2],X[t][k0+3]}
// B-frag mirrors A with cluster rows (B^T rows == centroids_weight rows).
__global__ __launch_bounds__(32)
void centroid_gemm_wmma(const float* __restrict__ X, const float* __restrict__ Cw,
                        float* __restrict__ logits, int T, int C) {
  const int lane  = threadIdx.x;          // 0..31, wave32
  const int cTile = blockIdx.x * 16;
  const int tTile = blockIdx.y * 16;
  const int rowA  = tTile + (lane & 15);
  const int rowB  = cTile + (lane & 15);
  const int kOff  = (lane >> 4) << 1;     // 0 or 2
  const float* aPtr = X  + (size_t)rowA * H_DIM + kOff;
  const float* bPtr = Cw + (size_t)rowB * H_DIM + kOff;

  v8f acc = {};
#if __has_builtin(__builtin_amdgcn_wmma_f32_16x16x4_f32)
  for (int k0 = 0; k0 < H_DIM; k0 += 4) {
    v2f a = *(const v2f*)(aPtr + k0);     // global_load_b64, 8B aligned
    v2f b = *(const v2f*)(bPtr + k0);
    acc = __builtin_amdgcn_wmma_f32_16x16x4_f32(
        /*neg_a=*/false, a, /*neg_b=*/false, b,
        /*c_mod=*/(short)0, acc, /*reuse_a=*/false, /*reuse_b=*/false);
  }
#else
  // scalar fallback with identical D layout
  {
    const int n     = cTile + (lane & 15);
    const int mBase = tTile + ((lane >> 4) << 3);
    for (int i = 0; i < 8; ++i) {
      float s = 0.f;
      const float* xr = X + (size_t)(mBase + i) * H_DIM;
      const float* cr = Cw + (size_t)n * H_DIM;
      for (int k = 0; k < H_DIM; ++k) s = fmaf(xr[k], cr[k], s);
      acc[i] = s;
    }
  }
#endif
  const int n     = cTile + (lane & 15);
  const int mBase = tTile + ((lane >> 4) << 3);   // lanes 0-15 -> M=i, 16-31 -> M=8+i
  for (int i = 0; i < 8; ++i)
    logits[(size_t)(mBase + i) * C + n] = acc[i];
}

// ---- Phase 2: top-2 clusters per token (jax.lax.top_k tie-break order) -----
__global__ void topk2_kernel(const float* __restrict__ logits, int2* __restrict__ top2,
                             int T, int C) {
  int t = blockIdx.x * blockDim.x + threadIdx.x;
  if (t >= T) return;
  const float* row = logits + (size_t)t * C;
  float b1 = -__int_as_float(0x7f800000), b2 = b1;
  int i1 = 0, i2 = 0;
  for (int c = 0; c < C; ++c) {
    float v = row[c];
    if (v > b1)      { b2 = b1; i2 = i1; b1 = v; i1 = c; }
    else if (v > b2) { b2 = v;  i2 = c; }
  }
  top2[t] = make_int2(i1, i2);
}

// ---- Phase 3: candidate logits (GEMV per token, wave-per-row dots) ---------
// One block (8 waves) per token. x staged in LDS; each wave handles candidate
// rows strided by 8. Lanes do coalesced float4 loads of the lm_head row.
__global__ __launch_bounds__(256)
void selected_logits_kernel(const float* __restrict__ X, const float* __restrict__ W,
                            const int* __restrict__ tord, const int2* __restrict__ top2,
                            float* __restrict__ sel, unsigned* __restrict__ minEnc,
                            int vpc) {
  __shared__ float xs[H_DIM];
  const int t = blockIdx.x;
  const float* x = X + (size_t)t * H_DIM;
  for (int i = threadIdx.x; i < H_DIM; i += blockDim.x) xs[i] = x[i];
  __syncthreads();

  const int2 cc   = top2[t];
  const int  wave = threadIdx.x >> 5;
  const int  lane = threadIdx.x & 31;
  const int  nC   = 2 * vpc;                    // candidates per token (512)
  const float4* xs4 = (const float4*)xs;

  float localMin = __int_as_float(0x7f800000);  // +inf
  for (int j = wave; j < nC; j += 8) {
    const int cluster = (j < vpc) ? cc.x : cc.y;
    const int jj      = (j < vpc) ? j : (j - vpc);
    const int row     = tord[cluster * vpc + jj];      // canonical vocab position
    const float4* wr  = (const float4*)(W + (size_t)row * H_DIM);
    float acc = 0.f;
    #pragma unroll
    for (int i = 0; i < H_DIM / 128; ++i) {            // 32 lanes * 4 floats
      float4 w4 = wr[lane + i * 32];
      float4 x4 = xs4[lane + i * 32];
      acc = fmaf(w4.x, x4.x, acc);
      acc = fmaf(w4.y, x4.y, acc);
      acc = fmaf(w4.z, x4.z, acc);
      acc = fmaf(w4.w, x4.w, acc);
    }
    #pragma unroll
    for (int off = 16; off > 0; off >>= 1) acc += __shfl_xor(acc, off, 32);
    if (lane == 0) {
      sel[(size_t)t * nC + j] = acc;
      localMin = fminf(localMin, acc);
    }
  }
  if (lane == 0) atomicMin(minEnc, fenc(localMin));
}

// ---- Phase 4: fill full vocab with (min - 1) (256 MB, b128 stores) ---------
__global__ void fill_mask_kernel(float4* __restrict__ out, const unsigned* __restrict__ minEnc,
                                 size_t n4) {
  float m = fdec(*minEnc) - 1.0f;
  size_t i = (size_t)blockIdx.x * blockDim.x + threadIdx.x;
  if (i < n4) out[i] = make_float4(m, m, m, m);
}

// ---- Phase 5: scatter candidate logits back into full vocab ----------------
__global__ void scatter_kernel(const float* __restrict__ sel, const int2* __restrict__ top2,
                               const int* __restrict__ tord, float* __restrict__ out,
                               int T, int V, int vpc) {
  const int nC  = 2 * vpc;
  int idx = blockIdx.x * blockDim.x + threadIdx.x;
  if (idx >= T * nC) return;
  const int t = idx / nC;
  const int j = idx - t * nC;
  const int2 cc     = top2[t];
  const int cluster = (j < vpc) ? cc.x : cc.y;
  const int jj      = (j < vpc) ? j : (j - vpc);
  const int pos     = tord[cluster * vpc + jj];
  out[(size_t)t * V + pos] = sel[idx];
}

extern "C" void kernel_launch(void* const* d_in, const int* in_sizes, int n_in,
                              void* d_out, int out_size, void* d_ws, size_t ws_size,
                              hipStream_t stream) {
  const float* X    = (const float*)d_in[0];   // hidden_states  [T, H]
  const float* W    = (const float*)d_in[1];   // lm_head_weight [V, H]
  const float* Cw   = (const float*)d_in[2];   // centroids      [C, H]
  const int*   tord = (const int*)d_in[3];     // token_ordering [V]
  float* out = (float*)d_out;

  const int T   = in_sizes[0] / H_DIM;   // 512 tokens (B*S)
  const int V   = in_sizes[1] / H_DIM;   // 131072
  const int C   = in_sizes[2] / H_DIM;   // 512
  const int vpc = V / C;                 // 256
  const int nC  = 2 * vpc;               // candidates per token

  // Workspace layout. `sel` aliases `logits` (logits dead after top-k).
  char* ws = (char*)d_ws;
  float*    logits = (float*)ws;                                        // T*C f32
  float*    sel    = (float*)ws;                                        // T*nC f32 (alias)
  size_t    big    = (size_t)T * ((C > nC) ? C : nC) * sizeof(float);
  int2*     top2   = (int2*)(ws + big);                                 // T int2
  unsigned* minEnc = (unsigned*)(ws + big + (size_t)T * sizeof(int2));  // 1 uint

  init_min_kernel<<<1, 1, 0, stream>>>(minEnc);

  centroid_gemm_wmma<<<dim3(C / 16, T / 16), 32, 0, stream>>>(X, Cw, logits, T, C);

  topk2_kernel<<<(T + 255) / 256, 256, 0, stream>>>(logits, top2, T, C);

  selected_logits_kernel<<<T, 256, 0, stream>>>(X, W, tord, top2, sel, minEnc, vpc);

  size_t n4 = (size_t)out_size / 4;
  fill_mask_kernel<<<(unsigned)((n4 + 255) / 256), 256, 0, stream>>>((float4*)out, minEnc, n4);

  scatter_kernel<<<(T * nC + 255) / 256, 256, 0, stream>>>(sel, top2, tord, out, T, V, vpc);
}